// DeepseekV4HyperConnection_18425409700454
// MI455X (gfx1250) — compile-verified
//
#include <hip/hip_runtime.h>
#include <math.h>

typedef __attribute__((ext_vector_type(2))) float v2f;
typedef __attribute__((ext_vector_type(8))) float v8f;

#define HC_     4
#define HIDDEN_ 4096
#define DIM_    (HC_ * HIDDEN_)     // 16384 floats per token
#define MIX_    24                  // (2+HC)*HC gate logits
#define TOKS_   16                  // tokens per workgroup (one WMMA M tile)
#define WAVES_  8                   // wave32s per workgroup
#define KSLICE_ (DIM_ / WAVES_)     // 2048 K per wave

__global__ __launch_bounds__(256) void hc_fused_kernel(
    const float* __restrict__ hs,     // [ntok, 16384] flattened hidden streams
    const float* __restrict__ fn,     // [24, 16384]
    const float* __restrict__ base,   // [24]
    const float* __restrict__ scale,  // [3]
    float* __restrict__ out,          // pre[ntok*4] | post[ntok*4] | comb[ntok*16]
    int ntok)
{
    __shared__ float lds_c0[WAVES_ * 256];   // per-wave 16x16 partials, cols 0..15
    __shared__ float lds_c1[WAVES_ * 128];   // per-wave 16x8  partials, cols 16..23
    __shared__ float lds_ss[WAVES_ * 32];    // per-wave per-lane sumsq partials
    __shared__ float lds_mix[TOKS_ * MIX_];  // reduced logits
    __shared__ float lds_rs[TOKS_];          // rsqrt(mean sq + eps)

    const int tid  = threadIdx.x;
    const int wave = tid >> 5;
    const int lane = tid & 31;
    const int half = lane >> 4;   // 0: K{0,1}, 1: K{2,3}
    const int l16  = lane & 15;
    const long tok0 = (long)blockIdx.x * TOKS_;

    // A fragment source: this lane's token row (16x4 f32 WMMA layout)
    const float* arow  = hs + (tok0 + l16) * (long)DIM_ + (half << 1);
    // B fragment sources: fn rows are the mix columns (4x16 K-major layout)
    const float* b0row = fn + (long)l16 * DIM_ + (half << 1);
    int n1 = 16 + l16; n1 = (n1 > 23) ? 23 : n1;  // clamp: cols 24..31 are dead padding
    const float* b1row = fn + (long)n1 * DIM_ + (half << 1);

    v8f c0 = {0.f, 0.f, 0.f, 0.f, 0.f, 0.f, 0.f, 0.f};
    v8f c1 = {0.f, 0.f, 0.f, 0.f, 0.f, 0.f, 0.f, 0.f};
    float ss = 0.f;

    const int kbase = wave * KSLICE_;
    for (int kb = 0; kb < KSLICE_; kb += 32) {
        // stream-ahead on the HBM-bound operand (lowers to global_prefetch_b8)
        __builtin_prefetch(arow + kbase + kb + 512, 0, 0);
#pragma unroll
        for (int kk = 0; kk < 32; kk += 4) {
            const int k = kbase + kb + kk;
            v2f a  = *(const v2f*)(arow  + k);
            v2f b0 = *(const v2f*)(b0row + k);
            v2f b1 = *(const v2f*)(b1row + k);
            ss = fmaf(a.x, a.x, ss);
            ss = fmaf(a.y, a.y, ss);
            // D = A(16x4 f32) x B(4x16 f32) + C, fp32 accumulate
            c0 = __builtin_amdgcn_wmma_f32_16x16x4_f32(false, a, false, b0,
                                                       (short)0, c0, false, false);
            c1 = __builtin_amdgcn_wmma_f32_16x16x4_f32(false, a, false, b1,
                                                       (short)0, c1, false, false);
        }
    }

    // Spill per-wave partials to LDS in canonical (token, col) layout.
    // C layout: VGPR i, lanes 0-15 -> (M=i, N=lane); lanes 16-31 -> (M=i+8, N=lane-16)
#pragma unroll
    for (int i = 0; i < 8; ++i) {
        const int M = i + (half << 3);
        lds_c0[wave * 256 + M * 16 + l16] = c0[i];
        if (l16 < 8) lds_c1[wave * 128 + M * 8 + l16] = c1[i];
    }
    lds_ss[wave * 32 + lane] = ss;
    __syncthreads();

    // Cross-wave reduction of the 16x24 logit tile and per-token sumsq.
    {
        float s = 0.f;
#pragma unroll
        for (int w = 0; w < WAVES_; ++w) s += lds_c0[w * 256 + tid];
        lds_mix[(tid >> 4) * MIX_ + (tid & 15)] = s;
    }
    if (tid < 128) {
        float s = 0.f;
#pragma unroll
        for (int w = 0; w < WAVES_; ++w) s += lds_c1[w * 128 + tid];
        lds_mix[(tid >> 3) * MIX_ + 16 + (tid & 7)] = s;
    }
    if (tid < TOKS_) {
        float s = 0.f;
#pragma unroll
        for (int w = 0; w < WAVES_; ++w)
            s += lds_ss[w * 32 + tid] + lds_ss[w * 32 + 16 + tid];
        lds_rs[tid] = rsqrtf(s * (1.0f / (float)DIM_) + 1e-6f);
    }
    __syncthreads();

    // Finalize: gates + softmax + Sinkhorn, one thread per token.
    if (tid < TOKS_) {
        const int t = tid;
        const float r = lds_rs[t];
        float m[MIX_];
#pragma unroll
        for (int i = 0; i < MIX_; ++i) m[i] = lds_mix[t * MIX_ + i] * r;

        const float s0 = scale[0], s1 = scale[1], s2 = scale[2];
        const long g = tok0 + t;
        float* outPre  = out;
        float* outPost = out + (long)ntok * 4;
        float* outComb = out + (long)ntok * 8;

#pragma unroll
        for (int i = 0; i < 4; ++i)
            outPre[g * 4 + i] = 1.f / (1.f + expf(-(m[i] * s0 + base[i]))) + 1e-5f;
#pragma unroll
        for (int i = 0; i < 4; ++i)
            outPost[g * 4 + i] = 2.f / (1.f + expf(-(m[4 + i] * s1 + base[4 + i])));

        float c[4][4];
#pragma unroll
        for (int i = 0; i < 4; ++i) {
            float mx = -3.4e38f;
#pragma unroll
            for (int j = 0; j < 4; ++j) {
                c[i][j] = m[8 + i * 4 + j] * s2 + base[8 + i * 4 + j];
                mx = fmaxf(mx, c[i][j]);
            }
            float sum = 0.f;
#pragma unroll
            for (int j = 0; j < 4; ++j) { c[i][j] = expf(c[i][j] - mx); sum += c[i][j]; }
            const float inv = 1.f / sum;
#pragma unroll
            for (int j = 0; j < 4; ++j) c[i][j] = c[i][j] * inv + 1e-5f;
        }
        // Sinkhorn: column-normalize, then 4x (row-normalize, column-normalize)
#pragma unroll
        for (int j = 0; j < 4; ++j) {
            const float cs = c[0][j] + c[1][j] + c[2][j] + c[3][j];
            const float inv = 1.f / (cs + 1e-5f);
#pragma unroll
            for (int i = 0; i < 4; ++i) c[i][j] *= inv;
        }
#pragma unroll
        for (int it = 0; it < 4; ++it) {
#pragma unroll
            for (int i = 0; i < 4; ++i) {
                const float rs = c[i][0] + c[i][1] + c[i][2] + c[i][3];
                const float inv = 1.f / (rs + 1e-5f);
#pragma unroll
                for (int j = 0; j < 4; ++j) c[i][j] *= inv;
            }
#pragma unroll
            for (int j = 0; j < 4; ++j) {
                const float cs = c[0][j] + c[1][j] + c[2][j] + c[3][j];
                const float inv = 1.f / (cs + 1e-5f);
#pragma unroll
                for (int i = 0; i < 4; ++i) c[i][j] *= inv;
            }
        }
#pragma unroll
        for (int i = 0; i < 4; ++i)
#pragma unroll
            for (int j = 0; j < 4; ++j)
                outComb[g * 16 + i * 4 + j] = c[i][j];
    }
}

extern "C" void kernel_launch(void* const* d_in, const int* in_sizes, int n_in,
                              void* d_out, int out_size, void* d_ws, size_t ws_size,
                              hipStream_t stream) {
    const float* hs    = (const float*)d_in[0];  // [B,S,HC,HIDDEN] fp32
    const float* fn    = (const float*)d_in[1];  // [24, 16384] fp32
    const float* base  = (const float*)d_in[2];  // [24]
    const float* scale = (const float*)d_in[3];  // [3]
    float* out = (float*)d_out;

    const int ntok = in_sizes[0] / DIM_;         // B*S = 8192
    dim3 grid(ntok / TOKS_);                     // 512 workgroups
    dim3 block(256);                             // 8 wave32s
    hipLaunchKernelGGL(hc_fused_kernel, grid, block, 0, stream,
                       hs, fn, base, scale, out, ntok);
}